// Concept_ood_24017457119454
// MI455X (gfx1250) — compile-verified
//
#include <hip/hip_runtime.h>
#include <hip/hip_bf16.h>
#include <math.h>

// ---------------------------------------------------------------------------
// Problem dims (fixed by setup_inputs): B=1024, S=512, E=1024, D=2048
// ---------------------------------------------------------------------------
#define DIM_B 1024
#define DIM_S 512
#define DIM_E 1024
#define DIM_D 2048
#define BN_EPS 1e-5f
#define SMOOTH 10.0f

// GEMM tiling: block = 128 threads (4 waves) -> 128x64 C tile
//   wave w owns rows [128*by + 32w, +32): 2 A fragments x 4 B tiles = 8 accums
#define MT        128                // block rows
#define KT        32                 // K chunk staged per LDS buffer
#define A_STRIDE  36                 // 32 + 4 pad floats: 144B rows (16B aligned, conflict-free)
#define B_STRIDE  68                 // 64 + 4 pad floats: 272B rows (16B aligned, conflict-free)

typedef float v2f __attribute__((ext_vector_type(2)));
typedef float v8f __attribute__((ext_vector_type(8)));

// ---------------------------------------------------------------------------
// CDNA5 async global->LDS staging (ASYNCcnt-tracked DMA into LDS).
// Builtin prototype (probe-confirmed): v4i32 AS1* src, v4i32 AS3* dst,
// imm offset, imm cpol.
// ---------------------------------------------------------------------------
typedef int async_v4i __attribute__((vector_size(16)));

#if __has_builtin(__builtin_amdgcn_global_load_async_to_lds_b128)
#define HAVE_ASYNC_LDS 1
#define ASYNC_LDS_B128(gp, lp)                                                 \
    __builtin_amdgcn_global_load_async_to_lds_b128(                            \
        (__attribute__((address_space(1))) async_v4i*)(gp),                    \
        (__attribute__((address_space(3))) async_v4i*)(lp), 0, 0)
#else
#define HAVE_ASYNC_LDS 0
#endif

template <int N>
__device__ __forceinline__ void wait_async_le() {
#if __has_builtin(__builtin_amdgcn_s_wait_asynccnt)
    __builtin_amdgcn_s_wait_asynccnt(N);
#else
    asm volatile("s_wait_asynccnt %0" :: "i"(N) : "memory");
#endif
}

// ---------------------------------------------------------------------------
// WMMA f32 GEMM:  C(MxN) = A(MxK) @ B(KxN or, if BT, stored NxK) + bias
//   bias_mode: 0 = none, 1 = per-column vector, 2 = scalar (bias[0])
//
// K loop staged through LDS in KT=32 chunks, double buffered:
//   async path:   global_load_async_to_lds_b128 streams chunk c+1 while the
//                 matrix pipe consumes chunk c (s_wait_asynccnt 12 / 0).
//   fallback:     float4 global loads -> regs -> ds_store_b128.
//
// LDS layouts (per buffer):
//   A panel:           ldsA[m][k]  m<128, k<KT, row stride A_STRIDE
//   B panel (BT=0):    ldsB[k][n]  k<KT,  n<64, row stride B_STRIDE
//   B panel (BT=1):    ldsB[n][k]  n<64,  k<KT, row stride A_STRIDE
//
// V_WMMA_F32_16X16X4_F32 operand layout (ISA 7.12.2):
//   A (16x4): lanes 0-15: row=lane, {v0,v1}={K0,K1}; lanes 16-31: {K2,K3}
//   B (4x16): lanes 0-15: col=lane, {v0,v1}={K0,K1}; lanes 16-31: {K2,K3}
//   C/D:      vgpr i, lanes 0-15 -> (M=i, N=lane); 16-31 -> (M=i+8, N=lane-16)
// ---------------------------------------------------------------------------
template <bool BT>
__global__ __launch_bounds__(128)
void wmma_gemm_f32(const float* __restrict__ A, const float* __restrict__ B,
                   const float* __restrict__ bias, int bias_mode,
                   float* __restrict__ C,
                   int K, int lda, int ldb, int ldc)
{
    __shared__ float ldsA[2][MT * A_STRIDE];
    __shared__ float ldsB[2][64 * A_STRIDE];   // sized for the larger (BT) layout

    const int tid  = threadIdx.x;
    const int lane = tid & 31;
    const int wave = tid >> 5;                 // 0..3
    const int half = lane >> 4;                // K-pair select
    const int l    = lane & 15;                // row (A) / col (B,C) in tile

    const int m0 = blockIdx.y * MT;
    const int n0 = blockIdx.x * 64;

    v8f acc[8] = {v8f{}, v8f{}, v8f{}, v8f{}, v8f{}, v8f{}, v8f{}, v8f{}};

    const int chunks = K / KT;

#if HAVE_ASYNC_LDS
    // ---- async staging: 8 A-chunks + 4 B-chunks of 16B per thread ----
    auto stage = [&](int buf, int k0) {
#pragma unroll
        for (int i = 0; i < 8; ++i) {          // A: 128 rows x 32 floats = 1024 x b128
            const int g = tid + i * 128;
            const int m = g >> 3, kc = (g & 7) << 2;
            ASYNC_LDS_B128(A + (size_t)(m0 + m) * lda + k0 + kc,
                           &ldsA[buf][m * A_STRIDE + kc]);
        }
#pragma unroll
        for (int i = 0; i < 4; ++i) {
            const int g = tid + i * 128;
            if (BT) {                           // rows of B (NxK) copied as-is
                const int n = g >> 3, kc = (g & 7) << 2;
                ASYNC_LDS_B128(B + (size_t)(n0 + n) * ldb + k0 + kc,
                               &ldsB[buf][n * A_STRIDE + kc]);
            } else {                            // rows of B (KxN) copied as-is
                const int k = g >> 4, nc = (g & 15) << 2;
                ASYNC_LDS_B128(B + (size_t)(k0 + k) * ldb + n0 + nc,
                               &ldsB[buf][k * B_STRIDE + nc]);
            }
        }
    };

    stage(0, 0);
    for (int c = 0; c < chunks; ++c) {
        if (c + 1 < chunks) {
            stage((c + 1) & 1, (c + 1) * KT);   // stream next chunk
            wait_async_le<12>();                // chunk c resident (12 newer in flight)
        } else {
            wait_async_le<0>();
        }
        __syncthreads();                        // all waves' DMA visible

        const float* Ab = ldsA[c & 1];
        const float* Bb = ldsB[c & 1];
#pragma unroll
        for (int k = 0; k < KT; k += 4) {
            v2f a0 = *(const v2f*)(Ab + (wave * 32 + l)      * A_STRIDE + k + 2 * half);
            v2f a1 = *(const v2f*)(Ab + (wave * 32 + 16 + l) * A_STRIDE + k + 2 * half);
#pragma unroll
            for (int t = 0; t < 4; ++t) {
                v2f b;
                if (BT) {
                    b = *(const v2f*)(Bb + (t * 16 + l) * A_STRIDE + k + 2 * half);
                } else {
                    const float* bp = Bb + (k + 2 * half) * B_STRIDE + t * 16 + l;
                    b.x = bp[0];
                    b.y = bp[B_STRIDE];
                }
                acc[t] = __builtin_amdgcn_wmma_f32_16x16x4_f32(
                    false, a0, false, b, (short)0, acc[t], false, false);
                acc[4 + t] = __builtin_amdgcn_wmma_f32_16x16x4_f32(
                    false, a1, false, b, (short)0, acc[4 + t], false, false);
            }
        }
        __syncthreads();                        // compute done before buf reuse
    }
#else
    // ---- fallback: register-prefetch double buffer ----
    float4 ra[8], rb[4];
    auto gload = [&](int k0) {
#pragma unroll
        for (int i = 0; i < 8; ++i) {
            const int g = tid + i * 128;
            const int m = g >> 3, kc = (g & 7) << 2;
            ra[i] = *(const float4*)(A + (size_t)(m0 + m) * lda + k0 + kc);
        }
#pragma unroll
        for (int i = 0; i < 4; ++i) {
            const int g = tid + i * 128;
            if (BT) {
                const int n = g >> 3, kc = (g & 7) << 2;
                rb[i] = *(const float4*)(B + (size_t)(n0 + n) * ldb + k0 + kc);
            } else {
                const int k = g >> 4, nc = (g & 15) << 2;
                rb[i] = *(const float4*)(B + (size_t)(k0 + k) * ldb + n0 + nc);
            }
        }
    };
    auto lstore = [&](int buf) {
#pragma unroll
        for (int i = 0; i < 8; ++i) {
            const int g = tid + i * 128;
            const int m = g >> 3, kc = (g & 7) << 2;
            *(float4*)(&ldsA[buf][m * A_STRIDE + kc]) = ra[i];
        }
#pragma unroll
        for (int i = 0; i < 4; ++i) {
            const int g = tid + i * 128;
            if (BT) {
                const int n = g >> 3, kc = (g & 7) << 2;
                *(float4*)(&ldsB[buf][n * A_STRIDE + kc]) = rb[i];
            } else {
                const int k = g >> 4, nc = (g & 15) << 2;
                *(float4*)(&ldsB[buf][k * B_STRIDE + nc]) = rb[i];
            }
        }
    };

    gload(0);
    for (int c = 0; c < chunks; ++c) {
        lstore(c & 1);
        if (c + 1 < chunks) gload((c + 1) * KT);   // overlap next global fetch
        __syncthreads();

        const float* Ab = ldsA[c & 1];
        const float* Bb = ldsB[c & 1];
#pragma unroll
        for (int k = 0; k < KT; k += 4) {
            v2f a0 = *(const v2f*)(Ab + (wave * 32 + l)      * A_STRIDE + k + 2 * half);
            v2f a1 = *(const v2f*)(Ab + (wave * 32 + 16 + l) * A_STRIDE + k + 2 * half);
#pragma unroll
            for (int t = 0; t < 4; ++t) {
                v2f b;
                if (BT) {
                    b = *(const v2f*)(Bb + (t * 16 + l) * A_STRIDE + k + 2 * half);
                } else {
                    const float* bp = Bb + (k + 2 * half) * B_STRIDE + t * 16 + l;
                    b.x = bp[0];
                    b.y = bp[B_STRIDE];
                }
                acc[t] = __builtin_amdgcn_wmma_f32_16x16x4_f32(
                    false, a0, false, b, (short)0, acc[t], false, false);
                acc[4 + t] = __builtin_amdgcn_wmma_f32_16x16x4_f32(
                    false, a1, false, b, (short)0, acc[4 + t], false, false);
            }
        }
        // stores for chunk c+1 target the other buffer; the barrier above
        // (reached only after every wave finished chunk c-1) protects reuse.
    }
#endif

#pragma unroll
    for (int p = 0; p < 2; ++p) {
#pragma unroll
        for (int t = 0; t < 4; ++t) {
            const int col = n0 + t * 16 + l;
            float badd = 0.0f;
            if (bias_mode == 1)      badd = bias[col];
            else if (bias_mode == 2) badd = bias[0];
#pragma unroll
            for (int i = 0; i < 8; ++i) {
                const int row = m0 + wave * 32 + p * 16 + i + half * 8;
                C[(size_t)row * ldc + col] = acc[p * 4 + t][i] + badd;
            }
        }
    }
}

// ---------------------------------------------------------------------------
// Column-wise batch statistics: mu[e], biased var[e] over M rows.
// ---------------------------------------------------------------------------
__global__ void bn_stats_kernel(const float* __restrict__ Z, int M, int E,
                                float* __restrict__ mu, float* __restrict__ var)
{
    const int e = blockIdx.x * blockDim.x + threadIdx.x;
    if (e >= E) return;
    float s = 0.0f, ss = 0.0f;
    for (int r = 0; r < M; ++r) {
        const float z = Z[(size_t)r * E + e];
        s  += z;
        ss += z * z;
    }
    const float m = s / (float)M;
    mu[e]  = m;
    var[e] = ss / (float)M - m * m;
}

// ---------------------------------------------------------------------------
// W = tanh(g*(Z - mu)*rsqrt(var+eps) + b) [* scale]   (in-place safe)
// ---------------------------------------------------------------------------
__global__ void bn_tanh_kernel(const float* __restrict__ Z, float* __restrict__ W,
                               const float* __restrict__ g, const float* __restrict__ b,
                               const float* __restrict__ mu, const float* __restrict__ var,
                               const float* __restrict__ scale, int total, int E)
{
    for (int idx = blockIdx.x * blockDim.x + threadIdx.x; idx < total;
         idx += gridDim.x * blockDim.x) {
        const int e = idx & (E - 1);   // E is a power of two (1024)
        const float inv = rsqrtf(var[e] + BN_EPS);
        float w = tanhf(g[e] * (Z[idx] - mu[e]) * inv + b[e]);
        if (scale) w *= scale[e];
        W[idx] = w;
    }
}

// ---------------------------------------------------------------------------
// inv[r] = 1 / (||X[r,:]||_2 + 1e-8)   (one block per row)
// ---------------------------------------------------------------------------
__global__ void row_invnorm_kernel(const float* __restrict__ X, int Dc,
                                   float* __restrict__ inv)
{
    __shared__ float red[256];
    const int r = blockIdx.x;
    float s = 0.0f;
    for (int d = threadIdx.x; d < Dc; d += blockDim.x) {
        const float x = X[(size_t)r * Dc + d];
        s += x * x;
    }
    red[threadIdx.x] = s;
    __syncthreads();
    for (int off = blockDim.x >> 1; off > 0; off >>= 1) {
        if ((int)threadIdx.x < off) red[threadIdx.x] += red[threadIdx.x + off];
        __syncthreads();
    }
    if (threadIdx.x == 0) inv[r] = 1.0f / (sqrtf(red[0]) + 1e-8f);
}

// ---------------------------------------------------------------------------
// Row softmax of (SMOOTH * a). Writes weights to Wout and the concept-norm
// scaled copy (w * invcf[s]) to SWout so the final GEMM needs no cf copy.
// ---------------------------------------------------------------------------
__global__ void softmax_scale_kernel(const float* __restrict__ A, int Cols,
                                     const float* __restrict__ invcf,
                                     float* __restrict__ Wout,
                                     float* __restrict__ SWout)
{
    __shared__ float red[256];
    const int b = blockIdx.x;
    const float* a = A + (size_t)b * Cols;
    float* wrow  = Wout  + (size_t)b * Cols;
    float* swrow = SWout + (size_t)b * Cols;

    float mx = -1e30f;
    for (int s = threadIdx.x; s < Cols; s += blockDim.x)
        mx = fmaxf(mx, a[s] * SMOOTH);
    red[threadIdx.x] = mx;
    __syncthreads();
    for (int off = blockDim.x >> 1; off > 0; off >>= 1) {
        if ((int)threadIdx.x < off)
            red[threadIdx.x] = fmaxf(red[threadIdx.x], red[threadIdx.x + off]);
        __syncthreads();
    }
    const float rowmax = red[0];
    __syncthreads();

    float sum = 0.0f;
    for (int s = threadIdx.x; s < Cols; s += blockDim.x) {
        const float e = __expf(a[s] * SMOOTH - rowmax);
        wrow[s] = e;
        sum += e;
    }
    red[threadIdx.x] = sum;
    __syncthreads();
    for (int off = blockDim.x >> 1; off > 0; off >>= 1) {
        if ((int)threadIdx.x < off) red[threadIdx.x] += red[threadIdx.x + off];
        __syncthreads();
    }
    const float rinv = 1.0f / red[0];
    __syncthreads();

    for (int s = threadIdx.x; s < Cols; s += blockDim.x) {
        const float w = wrow[s] * rinv;
        wrow[s]  = w;
        swrow[s] = w * invcf[s];
    }
}

// ---------------------------------------------------------------------------
// Y[r,:] = X[r,:] / (||X[r,:]|| + 1e-8)   (one block per row)
// ---------------------------------------------------------------------------
__global__ void l2norm_rows_kernel(const float* __restrict__ X,
                                   float* __restrict__ Y, int Dc)
{
    __shared__ float red[256];
    __shared__ float invs;
    const int r = blockIdx.x;
    float s = 0.0f;
    for (int d = threadIdx.x; d < Dc; d += blockDim.x) {
        const float x = X[(size_t)r * Dc + d];
        s += x * x;
    }
    red[threadIdx.x] = s;
    __syncthreads();
    for (int off = blockDim.x >> 1; off > 0; off >>= 1) {
        if ((int)threadIdx.x < off) red[threadIdx.x] += red[threadIdx.x + off];
        __syncthreads();
    }
    if (threadIdx.x == 0) invs = 1.0f / (sqrtf(red[0]) + 1e-8f);
    __syncthreads();
    const float iv = invs;
    for (int d = threadIdx.x; d < Dc; d += blockDim.x)
        Y[(size_t)r * Dc + d] = X[(size_t)r * Dc + d] * iv;
}

// ---------------------------------------------------------------------------
// Orchestration
// ---------------------------------------------------------------------------
extern "C" void kernel_launch(void* const* d_in, const int* in_sizes, int n_in,
                              void* d_out, int out_size, void* d_ws, size_t ws_size,
                              hipStream_t stream)
{
    (void)in_sizes; (void)n_in; (void)out_size; (void)ws_size;

    const float* emb_instance    = (const float*)d_in[0];   // (B,E)
    const float* concept_feature = (const float*)d_in[1];   // (S,D)
    const float* fc1_w = (const float*)d_in[3];             // (D,E)
    const float* fc1_b = (const float*)d_in[4];             // (E)
    const float* bn1_g = (const float*)d_in[5];
    const float* bn1_b = (const float*)d_in[6];
    const float* fc2_w = (const float*)d_in[7];             // (E,E)
    const float* fc2_b = (const float*)d_in[8];
    const float* bn2_g = (const float*)d_in[9];
    const float* bn2_b = (const float*)d_in[10];
    const float* fc3_w = (const float*)d_in[11];            // (E)
    const float* fc3_b = (const float*)d_in[12];            // (1)

    float* ws = (float*)d_ws;
    float* Z1    = ws;                                   // (S,E)  -> W_s in place
    float* Z2    = Z1 + (size_t)DIM_S * DIM_E;           // (B,E)  -> W_v*fc3_w in place
    float* a_s   = Z2 + (size_t)DIM_B * DIM_E;           // (B,S)
    float* sw    = a_s + (size_t)DIM_B * DIM_S;          // (B,S)  scaled weights
    float* Ybuf  = sw  + (size_t)DIM_B * DIM_S;          // (B,D)  pre-norm output
    float* mu1   = Ybuf + (size_t)DIM_B * DIM_D;         // (E)
    float* var1  = mu1  + DIM_E;                         // (E)
    float* mu2   = var1 + DIM_E;                         // (E)
    float* var2  = mu2  + DIM_E;                         // (E)
    float* invcf = var2 + DIM_E;                         // (S)

    float* out_emb = (float*)d_out;                               // (B,D)
    float* out_w   = out_emb + (size_t)DIM_B * DIM_D;             // (B,S)

    // 0) concept-feature row inverse norms
    row_invnorm_kernel<<<DIM_S, 256, 0, stream>>>(concept_feature, DIM_D, invcf);

    // 1) Z1 = cf @ fc1_w + fc1_b        (M=S, N=E, K=D)
    wmma_gemm_f32<false><<<dim3(DIM_E / 64, DIM_S / MT), 128, 0, stream>>>(
        concept_feature, fc1_w, fc1_b, 1, Z1, DIM_D, DIM_D, DIM_E, DIM_E);

    // 2) BN1 stats + W_s = tanh(BN(Z1))
    bn_stats_kernel<<<DIM_E / 256, 256, 0, stream>>>(Z1, DIM_S, DIM_E, mu1, var1);
    bn_tanh_kernel<<<(DIM_S * DIM_E) / 256, 256, 0, stream>>>(
        Z1, Z1, bn1_g, bn1_b, mu1, var1, nullptr, DIM_S * DIM_E, DIM_E);

    // 3) Z2 = emb @ fc2_w + fc2_b       (M=B, N=E, K=E)
    wmma_gemm_f32<false><<<dim3(DIM_E / 64, DIM_B / MT), 128, 0, stream>>>(
        emb_instance, fc2_w, fc2_b, 1, Z2, DIM_E, DIM_E, DIM_E, DIM_E);

    // 4) BN2 stats + W_v' = tanh(BN(Z2)) * fc3_w  (fc3_w fused here)
    bn_stats_kernel<<<DIM_E / 256, 256, 0, stream>>>(Z2, DIM_B, DIM_E, mu2, var2);
    bn_tanh_kernel<<<(DIM_B * DIM_E) / 256, 256, 0, stream>>>(
        Z2, Z2, bn2_g, bn2_b, mu2, var2, fc3_w, DIM_B * DIM_E, DIM_E);

    // 5) a_s = W_v' @ W_s^T + fc3_b     (M=B, N=S, K=E; B transposed)
    wmma_gemm_f32<true><<<dim3(DIM_S / 64, DIM_B / MT), 128, 0, stream>>>(
        Z2, Z1, fc3_b, 2, a_s, DIM_E, DIM_E, DIM_E, DIM_S);

    // 6) weights_v = softmax(10*a_s); sw = weights_v * invcf
    softmax_scale_kernel<<<DIM_B, 256, 0, stream>>>(a_s, DIM_S, invcf, out_w, sw);

    // 7) Y = sw @ concept_feature       (M=B, N=D, K=S)
    wmma_gemm_f32<false><<<dim3(DIM_D / 64, DIM_B / MT), 128, 0, stream>>>(
        sw, concept_feature, nullptr, 0, Ybuf, DIM_S, DIM_S, DIM_D, DIM_D);

    // 8) emb_concept = l2norm(Y)
    l2norm_rows_kernel<<<DIM_B, 256, 0, stream>>>(Ybuf, out_emb, DIM_D);
}